// torchDTW_51977694216605
// MI455X (gfx1250) — compile-verified
//
#include <hip/hip_runtime.h>
#include <math.h>

// ---------------- problem constants ----------------
#define INF_VAL 99999.0f
#define NN   6144           // N = M
#define MD   6145           // table row stride (M+1)
#define ROWS 512            // tile height
#define COLS 512            // tile width
#define CPT  4              // columns per thread
#define TPB  128            // threads per block (4 waves, wave32)
#define NBLK (NN / ROWS)    // 12 tiles per side

#define GLOBAL_AS __attribute__((address_space(1)))
#define LDS_AS    __attribute__((address_space(3)))

#if defined(__has_builtin)
#  if __has_builtin(__builtin_amdgcn_global_load_async_to_lds_b32)
#    define HAVE_ASYNC_LDS 1
#  else
#    define HAVE_ASYNC_LDS 0
#  endif
#else
#  define HAVE_ASYNC_LDS 0
#endif

// Write raw DP boundaries: row 0 = [0, INF...], col 0 = INF. Re-run every call
// so the in-place sqrt at the end of the previous call is undone deterministically.
__global__ void dtw_init(float* __restrict__ t) {
    int total = MD + NN;   // 6145 row-0 entries + 6144 col-0 entries (rows 1..N)
    for (int k = blockIdx.x * blockDim.x + threadIdx.x; k < total;
         k += gridDim.x * blockDim.x) {
        if (k < MD) {
            t[k] = (k == 0) ? 0.0f : INF_VAL;
        } else {
            long long r = (long long)(k - MD + 1);
            t[r * (long long)MD] = INF_VAL;
        }
    }
}

// One 512x512 tile per workgroup; tiles on anti-diagonal d run in parallel.
// Thread j owns 4 consecutive columns; thread-skewed wavefront over rows.
__global__ __launch_bounds__(TPB) void dtw_tile(float* __restrict__ tab,
                                                const float* __restrict__ x,
                                                const float* __restrict__ y,
                                                int d) {
    __shared__ float xs[ROWS];        // x values for this tile's rows
    __shared__ float Lc[ROWS + 1];    // boundary column left of tile (incl. corner)
    __shared__ float Tp[COLS + 1];    // boundary row above tile (incl. corner)
    __shared__ float buf[3][TPB];     // 3-slot rotating relay: last-col value per thread

    const int j    = threadIdx.x;
    const int tmin = (d < NBLK) ? d : (NBLK - 1);
    const int bi   = tmin - (int)blockIdx.x;
    const int bj   = d - bi;
    const long long rowbase = (long long)bi * ROWS;   // table row of top boundary
    const long long colbase = (long long)bj * COLS;   // table col of left boundary

    // Stage x rows into LDS via the CDNA5 async global->LDS path when available.
#if HAVE_ASYNC_LDS
    for (int r = j; r < ROWS; r += TPB) {
        __builtin_amdgcn_global_load_async_to_lds_b32(
            (GLOBAL_AS int*)(x + rowbase + r), (LDS_AS int*)&xs[r], 0, 0);
    }
#else
    for (int r = j; r < ROWS; r += TPB) xs[r] = x[rowbase + r];
#endif
    for (int r = j; r <= ROWS; r += TPB)
        Lc[r] = tab[(rowbase + r) * (long long)MD + colbase];
    for (int c = j; c <= COLS; c += TPB)
        Tp[c] = tab[rowbase * (long long)MD + colbase + c];
#if HAVE_ASYNC_LDS
    asm volatile("s_wait_asynccnt 0" ::: "memory");
#endif
    __syncthreads();

    const long long cb = colbase + (long long)CPT * j;   // first global col (inner idx)
    const float y0 = y[cb + 0], y1 = y[cb + 1], y2 = y[cb + 2], y3 = y[cb + 3];
    // "up" values seeded from the boundary row above the tile.
    float u0 = Tp[CPT * j + 1], u1 = Tp[CPT * j + 2],
          u2 = Tp[CPT * j + 3], u3 = Tp[CPT * j + 4];

    int cur = 0, prev = 2, pprev = 1;   // cur = t%3, prev = (t-1)%3, pprev = (t-2)%3
    for (int t = 0; t < ROWS + TPB - 1; ++t) {
        const int i = t - j;            // tile-local row this thread processes now
        if (i >= 0 && i < ROWS) {
            const float xi = xs[i];
            float leftN, diagN;
            if (j == 0) {               // left boundary of the tile
                leftN = Lc[i + 1];
                diagN = Lc[i];
            } else {                    // neighbor thread's last column
                leftN = buf[prev][j - 1];                     // its row i   (step t-1)
                diagN = (i == 0) ? Tp[CPT * j]                // boundary row
                                 : buf[pprev][j - 1];         // its row i-1 (step t-2)
            }
            float e0 = xi - y0; float v0 = e0 * e0 + fminf(u0, fminf(leftN, diagN));
            float e1 = xi - y1; float v1 = e1 * e1 + fminf(u1, fminf(v0, u0));
            float e2 = xi - y2; float v2 = e2 * e2 + fminf(u2, fminf(v1, u1));
            float e3 = xi - y3; float v3 = e3 * e3 + fminf(u3, fminf(v2, u2));

            // Scatter store: table fits in the 192MB global L2, so this stays on-chip.
            const long long o = (rowbase + i + 1) * (long long)MD + cb + 1;
            tab[o]     = v0;
            tab[o + 1] = v1;
            tab[o + 2] = v2;
            tab[o + 3] = v3;

            buf[cur][j] = v3;           // relay last column to thread j+1
            u0 = v0; u1 = v1; u2 = v2; u3 = v3;
        }
        __syncthreads();                // single barrier/step (3-slot buffer => no race)
        const int tmp = pprev; pprev = prev; prev = cur; cur = tmp;
    }
}

// Final elementwise sqrt over the whole table (streams L2 -> HBM once).
__global__ void dtw_sqrt(float* __restrict__ p, long long n) {
    const long long stride = (long long)gridDim.x * blockDim.x;
    const long long idx    = (long long)blockIdx.x * blockDim.x + threadIdx.x;
    const long long n4     = n >> 2;
    float4* p4 = (float4*)p;
    for (long long k = idx; k < n4; k += stride) {
        float4 v = p4[k];
        v.x = sqrtf(v.x); v.y = sqrtf(v.y); v.z = sqrtf(v.z); v.w = sqrtf(v.w);
        p4[k] = v;
    }
    for (long long k = (n4 << 2) + idx; k < n; k += stride)
        p[k] = sqrtf(p[k]);
}

extern "C" void kernel_launch(void* const* d_in, const int* in_sizes, int n_in,
                              void* d_out, int out_size, void* d_ws, size_t ws_size,
                              hipStream_t stream) {
    const float* x = (const float*)d_in[0];
    const float* y = (const float*)d_in[1];
    float* tab = (float*)d_out;

    // 1) raw boundaries (also resets the in-place sqrt from the previous call)
    dtw_init<<<48, 256, 0, stream>>>(tab);

    // 2) 23 tile-anti-diagonals; tiles within a diagonal are independent blocks
    for (int d = 0; d < 2 * NBLK - 1; ++d) {
        const int cnt = (d < NBLK) ? (d + 1) : (2 * NBLK - 1 - d);
        dtw_tile<<<cnt, TPB, 0, stream>>>(tab, x, y, d);
    }

    // 3) in-place sqrt of the full (N+1)x(M+1) table
    dtw_sqrt<<<2048, 256, 0, stream>>>(tab, (long long)out_size);
}